// ProposalGenerator_88545045774628
// MI455X (gfx1250) — compile-verified
//
#include <hip/hip_runtime.h>
#include <stdint.h>

// ---------------- problem constants (match reference) ----------------
#define T_LEN     16384
#define NLEN      8          // lengths are 8 << li, li = 0..7  (8..1024)
#define MAXP      100
#define MIN_SC    0.1f
#define NTHREADS  256

// ---------------- algorithm tuning ----------------
#define TILE      2048       // start positions per LDS tile (double-buffered)
#define HALO      1024       // max proposal length (end halo)
#define ETILE     (TILE + HALO)
#define NTILES    (T_LEN / TILE)
#define NBUCK     1024       // histogram bins over score float bits
#define BUCK_BASE 15744u     // 0x3D80: score>=0.1 -> (bits>>16) >= 0x3DCC
#define CAP       512        // collected-candidate cap per batch
#define USE_ASYNC_LDS 1      // CDNA5 async global->LDS path
// per-thread async ops per tile: TILE/(256*4)=2 start + ETILE/(256*4)=3 end
#define OPS_PER_TILE 5

// ---- CDNA5 async global->LDS copy of 16 bytes (ASYNCcnt-tracked) ----
__device__ __forceinline__ void async_copy16(void* lds_dst, const void* gsrc) {
#if USE_ASYNC_LDS
  unsigned lds = (unsigned)(uintptr_t)lds_dst;            // wave-relative LDS byte addr
  unsigned long long ga = (unsigned long long)(uintptr_t)gsrc;
  asm volatile("global_load_async_to_lds_b128 %0, %1, off"
               :: "v"(lds), "v"(ga) : "memory");
#else
  *(float4*)lds_dst = *(const float4*)gsrc;
#endif
}

// wait until at most N async ops remain outstanding (in-order completion)
template <int N>
__device__ __forceinline__ void wait_async_le() {
#if USE_ASYNC_LDS
  asm volatile("s_wait_asynccnt %0" :: "n"(N) : "memory");
#endif
}

__global__ __launch_bounds__(NTHREADS)
void ProposalGenerator_88545045774628_kernel(const float* __restrict__ sp,
                                             const float* __restrict__ ep,
                                             float* __restrict__ out)
{
  __shared__ float s_start[2][TILE];              // 16 KB
  __shared__ float s_end[2][ETILE];               // 24 KB
  __shared__ unsigned s_hist[NBUCK];              //  4 KB
  __shared__ unsigned long long s_keys[CAP];      //  4 KB
  __shared__ unsigned s_cnt;
  __shared__ unsigned s_cut;

  const int b   = blockIdx.x;
  const int tid = threadIdx.x;
  const float* __restrict__ srow = sp + (size_t)b * T_LEN;
  const float* __restrict__ erow = ep + (size_t)b * T_LEN;

  // Issue one tile's async loads (NO wait here — pipelined by caller).
  // Op count is uniform across threads (required for partial ASYNCcnt waits):
  // the end-halo address is clamped instead of skipped; the clamped region
  // corresponds to e >= T candidates which the compute loop never reads.
  auto issue_tile = [&](int buf, int t0) {
    for (int i = tid * 4; i < TILE; i += NTHREADS * 4)
      async_copy16(&s_start[buf][i], srow + t0 + i);
    for (int i = tid * 4; i < ETILE; i += NTHREADS * 4) {
      int p = t0 + i;                              // 16B-aligned group
      int pc = (p <= T_LEN - 4) ? p : (T_LEN - 4); // clamp, never OOB
      async_copy16(&s_end[buf][i], erow + pc);
    }
  };

  for (int i = tid; i < NBUCK; i += NTHREADS) s_hist[i] = 0u;
  if (tid == 0) { s_cnt = 0u; s_cut = 0u; }
  __syncthreads();

  unsigned cutoff = 0;

  // ===== two pipelined passes: 0 = histogram, 1 = collect >= cutoff =====
  for (int pass = 0; pass < 2; ++pass) {
    issue_tile(0, 0);                              // prologue: tile 0 in flight
    for (int tile = 0; tile < NTILES; ++tile) {
      const int buf = tile & 1;
      const int t0  = tile * TILE;
      if (tile + 1 < NTILES) {
        issue_tile(buf ^ 1, t0 + TILE);            // prefetch next tile
        wait_async_le<OPS_PER_TILE>();             // only current tile must land
      } else {
        wait_async_le<0>();
      }
      __syncthreads();

      for (int sl = tid; sl < TILE; sl += NTHREADS) {
        const float sv = s_start[buf][sl];
        const int sg = t0 + sl;
        #pragma unroll
        for (int li = 0; li < NLEN; ++li) {
          const int L = 8 << li;
          if (sg + L - 1 < T_LEN) {
            float sc = sqrtf(sv * s_end[buf][sl + L - 1]);
            if (sc >= MIN_SC) {
              unsigned bits = __float_as_uint(sc);
              unsigned bk = (bits >> 16) - BUCK_BASE;
              bk = bk < (NBUCK - 1u) ? bk : (NBUCK - 1u);
              if (pass == 0) {
                atomicAdd(&s_hist[bk], 1u);
              } else if (bk >= cutoff) {
                unsigned pos = atomicAdd(&s_cnt, 1u);
                if (pos < CAP) {
                  unsigned flat = (unsigned)(li * T_LEN + sg);
                  // value-descending, index-ascending tiebreak (matches top_k)
                  s_keys[pos] = ((unsigned long long)bits << 32) | (unsigned)(~flat);
                }
              }
            }
          }
        }
      }
      __syncthreads();   // compute done before this buffer is refilled
    }

    if (pass == 0) {
      // cutoff bin containing the 100th-largest score
      if (tid == 0) {
        unsigned cum = 0, cut = 0;
        for (int i = NBUCK - 1; i >= 0; --i) {
          cum += s_hist[i];
          if (cum >= (unsigned)MAXP) { cut = (unsigned)i; break; }
        }
        s_cut = cut;
        s_cnt = 0u;
      }
      __syncthreads();
      cutoff = s_cut;
    }
  }

  // =============== pad + bitonic sort (descending) ===============
  unsigned cnt = s_cnt; cnt = cnt < (unsigned)CAP ? cnt : (unsigned)CAP;
  for (int i = tid; i < CAP; i += NTHREADS)
    if ((unsigned)i >= cnt) s_keys[i] = 0ull;      // 0 sorts below all real keys
  __syncthreads();

  for (unsigned k = 2; k <= (unsigned)CAP; k <<= 1) {
    for (unsigned j = k >> 1; j > 0; j >>= 1) {
      for (unsigned i = tid; i < (unsigned)CAP; i += NTHREADS) {
        unsigned ij = i ^ j;
        if (ij > i) {
          unsigned long long a = s_keys[i], c = s_keys[ij];
          bool desc = ((i & k) == 0);
          if (desc ? (a < c) : (a > c)) { s_keys[i] = c; s_keys[ij] = a; }
        }
      }
      __syncthreads();
    }
  }

  // =============== emit top-100 rows [start, end, score] ===============
  if (tid < MAXP) {
    unsigned long long key = s_keys[tid];
    float o0 = 0.f, o1 = (float)T_LEN, o2 = 0.f;
    if (key != 0ull) {
      unsigned bits = (unsigned)(key >> 32);
      unsigned flat = ~((unsigned)(key & 0xFFFFFFFFu));
      unsigned si = flat & (T_LEN - 1u);
      unsigned li = flat >> 14;                    // /T_LEN
      o0 = (float)si;
      o1 = (float)(si + (8u << li));
      o2 = __uint_as_float(bits);
    }
    float* o = out + ((size_t)b * MAXP + tid) * 3;
    o[0] = o0; o[1] = o1; o[2] = o2;
  }
}

extern "C" void kernel_launch(void* const* d_in, const int* in_sizes, int n_in,
                              void* d_out, int out_size, void* d_ws, size_t ws_size,
                              hipStream_t stream) {
  const float* start_probs = (const float*)d_in[0];
  const float* end_probs   = (const float*)d_in[1];
  float* out = (float*)d_out;
  int B = in_sizes[0] / T_LEN;
  ProposalGenerator_88545045774628_kernel<<<dim3(B), dim3(NTHREADS), 0, stream>>>(
      start_probs, end_probs, out);
}